// GraphAttentionLayer_6262062317608
// MI455X (gfx1250) — compile-verified
//
#include <hip/hip_runtime.h>
#include <math.h>

#define N_NODES  50000
#define IN_DIM   128
#define OUT_DIM  128
#define H_HEADS  4
#define C_CH     32
#define E_EDGES  1600000
#define EE_TOTAL (E_EDGES + N_NODES)
#define NEG_SLOPE 0.2f
#define LN_EPS    1e-5f

typedef __attribute__((ext_vector_type(2))) float v2f;
typedef __attribute__((ext_vector_type(8))) float v8f;

// Order-preserving float -> uint map for atomicMax-based segment max.
__device__ __forceinline__ unsigned int fmap(float f) {
    unsigned int u = __float_as_uint(f);
    return (u & 0x80000000u) ? ~u : (u | 0x80000000u);
}
__device__ __forceinline__ float funmap(unsigned int u) {
    unsigned int b = (u & 0x80000000u) ? (u & 0x7FFFFFFFu) : ~u;
    return __uint_as_float(b);
}

// ---------------- GEMM: xp = x @ W  (fp32 WMMA, 16x16 tile per wave) ----------
__global__ void gemm_xp_wmma(const float* __restrict__ x,
                             const float* __restrict__ W,
                             float* __restrict__ xp) {
    const int mt   = blockIdx.x;            // 3125 M-tiles (50000/16)
    const int nt   = threadIdx.x >> 5;      // 8 waves -> 8 N-tiles (128/16)
    const int lane = threadIdx.x & 31;
    const int row  = lane & 15;             // A: M index, B/D: N index
    const int hi   = lane >> 4;             // lane half selects K pair
    const int khalf = hi << 1;              // 0 or 2

    const float* xrow = x + (mt * 16 + row) * IN_DIM;
    const float* wcol = W + nt * 16 + row;  // column of W (N = lane&15)

    v8f c = {};
#pragma unroll
    for (int kk = 0; kk < IN_DIM / 4; ++kk) {
        const int k0 = kk * 4 + khalf;
        v2f a; a.x = xrow[k0];               a.y = xrow[k0 + 1];
        v2f b; b.x = wcol[k0 * OUT_DIM];     b.y = wcol[(k0 + 1) * OUT_DIM];
        c = __builtin_amdgcn_wmma_f32_16x16x4_f32(
                /*neg_a=*/false, a, /*neg_b=*/false, b,
                /*c_mod=*/(short)0, c, /*reuse_a=*/false, /*reuse_b=*/false);
    }
    // C/D layout: VGPR r, lanes 0-15 -> M=r, lanes 16-31 -> M=r+8, N=lane&15
    const int mbase = mt * 16 + (hi << 3);
    const int col   = nt * 16 + row;
#pragma unroll
    for (int r = 0; r < 8; ++r)
        xp[(mbase + r) * OUT_DIM + col] = c[r];
}

// ------------- per-(node,head) attention dot products (one wave each) ---------
__global__ void node_attn_dots(const float* __restrict__ xp,
                               const float* __restrict__ att_src,
                               const float* __restrict__ att_dst,
                               float* __restrict__ a_src,
                               float* __restrict__ a_dst) {
    const int wave = (blockIdx.x * blockDim.x + threadIdx.x) >> 5;
    const int lane = threadIdx.x & 31;
    if (wave >= N_NODES * H_HEADS) return;
    const int n = wave >> 2, h = wave & 3;
    const float v = xp[n * OUT_DIM + h * C_CH + lane];
    float s = v * att_src[h * C_CH + lane];
    float d = v * att_dst[h * C_CH + lane];
    for (int off = 16; off > 0; off >>= 1) {
        s += __shfl_xor(s, off, 32);
        d += __shfl_xor(d, off, 32);
    }
    if (lane == 0) { a_src[n * 4 + h] = s; a_dst[n * 4 + h] = d; }
}

// -------- we8[h] = dot(W_edge[h*32..], att_edge[h*32..])  (4 waves) ----------
__global__ void we8_kernel(const float* __restrict__ W_edge,
                           const float* __restrict__ att_edge,
                           float* __restrict__ we8) {
    const int h = threadIdx.x >> 5;
    const int lane = threadIdx.x & 31;
    float p = W_edge[h * 32 + lane] * att_edge[h * 32 + lane];
    for (int off = 16; off > 0; off >>= 1) p += __shfl_xor(p, off, 32);
    if (lane == 0) we8[h] = p;
}

// ------------------ sum(edge_weight) for self-loop fill value ----------------
__global__ void ewsum_kernel(const float* __restrict__ ew, float* __restrict__ out) {
    __shared__ float sh[256];
    const int i = blockIdx.x * blockDim.x + threadIdx.x;
    sh[threadIdx.x] = (i < E_EDGES) ? ew[i] : 0.f;
    __syncthreads();
    for (int s = 128; s > 0; s >>= 1) {
        if (threadIdx.x < s) sh[threadIdx.x] += sh[threadIdx.x + s];
        __syncthreads();
    }
    if (threadIdx.x == 0) atomicAdd(out, sh[0]);
}

__device__ __forceinline__ void edge_decode(int e, const long long* ei,
                                            const float* ew, const float* ewsum,
                                            int& s, int& d, float& w) {
    if (e < E_EDGES) {
        s = (int)ei[e]; d = (int)ei[E_EDGES + e]; w = ew[e];
    } else {
        s = d = e - E_EDGES; w = ewsum[0] * (1.0f / (float)E_EDGES);
    }
}

// ------------------------- pass 1: segment max -------------------------------
__global__ void edge_max_kernel(const long long* __restrict__ ei,
                                const float* __restrict__ ew,
                                const float* __restrict__ a_src,
                                const float* __restrict__ a_dst,
                                const float* __restrict__ we8,
                                const float* __restrict__ ewsum,
                                unsigned int* __restrict__ mmax) {
    const int e = blockIdx.x * blockDim.x + threadIdx.x;
    if (e >= EE_TOTAL) return;
    int s, d; float w;
    edge_decode(e, ei, ew, ewsum, s, d, w);
#pragma unroll
    for (int h = 0; h < 4; ++h) {
        float l = a_src[s * 4 + h] + a_dst[d * 4 + h] + w * we8[h];
        l = (l > 0.f) ? l : NEG_SLOPE * l;
        atomicMax(&mmax[d * 4 + h], fmap(l));
    }
}

// ------------------------- pass 2: sum of exp --------------------------------
__global__ void edge_denom_kernel(const long long* __restrict__ ei,
                                  const float* __restrict__ ew,
                                  const float* __restrict__ a_src,
                                  const float* __restrict__ a_dst,
                                  const float* __restrict__ we8,
                                  const float* __restrict__ ewsum,
                                  const unsigned int* __restrict__ mmax,
                                  float* __restrict__ denom) {
    const int e = blockIdx.x * blockDim.x + threadIdx.x;
    if (e >= EE_TOTAL) return;
    int s, d; float w;
    edge_decode(e, ei, ew, ewsum, s, d, w);
#pragma unroll
    for (int h = 0; h < 4; ++h) {
        float l = a_src[s * 4 + h] + a_dst[d * 4 + h] + w * we8[h];
        l = (l > 0.f) ? l : NEG_SLOPE * l;
        atomicAdd(&denom[d * 4 + h], expf(l - funmap(mmax[d * 4 + h])));
    }
}

// ---------------- pass 3: weighted message scatter (wave per edge) -----------
__global__ void edge_scatter_kernel(const long long* __restrict__ ei,
                                    const float* __restrict__ ew,
                                    const float* __restrict__ a_src,
                                    const float* __restrict__ a_dst,
                                    const float* __restrict__ we8,
                                    const float* __restrict__ ewsum,
                                    const unsigned int* __restrict__ mmax,
                                    const float* __restrict__ denom,
                                    const float* __restrict__ xp,
                                    float* __restrict__ acc) {
    const int e    = (blockIdx.x * blockDim.x + threadIdx.x) >> 5;
    const int lane = threadIdx.x & 31;
    if (e >= EE_TOTAL) return;
    int s, d; float w;
    edge_decode(e, ei, ew, ewsum, s, d, w);
    float attn = 0.f;
    if (lane < 4) {   // one lane per head computes attn
        float l = a_src[s * 4 + lane] + a_dst[d * 4 + lane] + w * we8[lane];
        l = (l > 0.f) ? l : NEG_SLOPE * l;
        float ex = expf(l - funmap(mmax[d * 4 + lane]));
        attn = ex / (denom[d * 4 + lane] + 1e-16f);
    }
#pragma unroll
    for (int g = 0; g < 4; ++g) {            // channel group g == head g
        float ag = __shfl(attn, g, 32);
        int j = g * 32 + lane;               // coalesced across the wave
        atomicAdd(&acc[d * OUT_DIM + j], ag * xp[s * OUT_DIM + j]);
    }
}

// ------------------ epilogue: bias + LayerNorm + ELU -------------------------
__global__ void ln_elu_kernel(const float* __restrict__ acc,
                              const float* __restrict__ bias,
                              const float* __restrict__ gamma,
                              const float* __restrict__ beta,
                              float* __restrict__ out) {
    __shared__ float sh[128];
    const int n = blockIdx.x, t = threadIdx.x;
    float v = acc[n * OUT_DIM + t] + bias[t];
    sh[t] = v; __syncthreads();
    for (int s = 64; s > 0; s >>= 1) { if (t < s) sh[t] += sh[t + s]; __syncthreads(); }
    const float mu = sh[0] * (1.0f / OUT_DIM);
    __syncthreads();
    const float dv = v - mu;
    sh[t] = dv * dv; __syncthreads();
    for (int s = 64; s > 0; s >>= 1) { if (t < s) sh[t] += sh[t + s]; __syncthreads(); }
    const float var = sh[0] * (1.0f / OUT_DIM);
    float y = dv * rsqrtf(var + LN_EPS) * gamma[t] + beta[t];
    out[n * OUT_DIM + t] = (y > 0.f) ? y : (expf(y) - 1.0f);
}

extern "C" void kernel_launch(void* const* d_in, const int* in_sizes, int n_in,
                              void* d_out, int out_size, void* d_ws, size_t ws_size,
                              hipStream_t stream) {
    const float*     x        = (const float*)d_in[0];
    const long long* ei       = (const long long*)d_in[1];   // int64 [2,E]
    const float*     ew       = (const float*)d_in[2];
    const float*     W        = (const float*)d_in[3];
    const float*     att_src  = (const float*)d_in[4];
    const float*     att_dst  = (const float*)d_in[5];
    const float*     W_edge   = (const float*)d_in[6];
    const float*     att_edge = (const float*)d_in[7];
    const float*     bias     = (const float*)d_in[8];
    const float*     gamma    = (const float*)d_in[9];
    const float*     beta     = (const float*)d_in[10];
    float*           out      = (float*)d_out;

    float* ws = (float*)d_ws;
    float*        xp    = ws;                                   //  6,400,000
    float*        acc   = ws + 6400000;                         //  6,400,000
    float*        denom = ws + 12800000;                        //    200,000
    unsigned int* mmax  = (unsigned int*)(ws + 13000000);       //    200,000
    float*        ewsum = ws + 13200000;                        //   64 (pad)
    float*        we8   = ws + 13200064;                        //   64 (pad)
    float*        asrc  = ws + 13200128;                        //    200,000
    float*        adst  = ws + 13400128;                        //    200,000

    // Zero accumulators every launch (acc | denom | mmax | ewsum contiguous).
    hipMemsetAsync(acc, 0, (size_t)(6400000 + 200000 + 200000 + 64) * sizeof(float), stream);

    we8_kernel<<<1, 128, 0, stream>>>(W_edge, att_edge, we8);
    ewsum_kernel<<<(E_EDGES + 255) / 256, 256, 0, stream>>>(ew, ewsum);
    gemm_xp_wmma<<<N_NODES / 16, 256, 0, stream>>>(x, W, xp);
    node_attn_dots<<<(N_NODES * H_HEADS) / 8, 256, 0, stream>>>(xp, att_src, att_dst, asrc, adst);
    edge_max_kernel<<<(EE_TOTAL + 255) / 256, 256, 0, stream>>>(ei, ew, asrc, adst, we8, ewsum, mmax);
    edge_denom_kernel<<<(EE_TOTAL + 255) / 256, 256, 0, stream>>>(ei, ew, asrc, adst, we8, ewsum, mmax, denom);
    edge_scatter_kernel<<<((size_t)EE_TOTAL * 32 + 255) / 256, 256, 0, stream>>>(
        ei, ew, asrc, adst, we8, ewsum, mmax, denom, xp, acc);
    ln_elu_kernel<<<N_NODES, 128, 0, stream>>>(acc, bias, gamma, beta, out);
}